// Loss_17695265260053
// MI455X (gfx1250) — compile-verified
//
#include <hip/hip_runtime.h>
#include <math.h>

typedef __attribute__((ext_vector_type(2))) float v2f;
typedef __attribute__((ext_vector_type(8))) float v8f;

#define CHAMFER_EPS 1e-12f

// ---------------------------------------------------------------------------
// Kernel 1: one wave (32 threads) per (batch b, pred tile of 16 rows).
//   sym_ws[b][tile]  = sum over 16 rows n of  min_m sqrt(max(|p_n - t_m|^2, EPS))
//   asym_ws[b][tile] = sum over 16 rows n of  |p_n - t_n|
// Pairwise term via V_WMMA_F32_16X16X4_F32 with K-augmentation:
//   A[n,:] = [px,py,pz,1],  B[:,m] = [-2tx,-2ty,-2tz,|t|^2]
//   D[n,m] = -2 p.t + |t|^2 ;  |p|^2 added once after the min-reduction.
// ---------------------------------------------------------------------------
__global__ __launch_bounds__(32)
void chamfer_tile_kernel(const float* __restrict__ pred,
                         const float* __restrict__ targ,
                         float* __restrict__ sym_ws,
                         float* __restrict__ asym_ws,
                         int N, int nt)
{
    const int tile = blockIdx.x;          // pred tile (16 rows)
    const int b    = blockIdx.y;          // batch
    const int lane = threadIdx.x;         // 0..31 (wave32)
    const int li   = lane & 15;           // column / row-in-half index
    const int hi   = lane >> 4;           // half-wave select (K pair)

    const float* __restrict__ pb = pred + (size_t)b * N * 3;
    const float* __restrict__ tb = targ + (size_t)b * N * 3;

    // ---- A fragment (16x4 f32): lanes 0-15 -> K=0,1 ; lanes 16-31 -> K=2,3
    const int my_row = tile * 16 + li;
    const float px = pb[(size_t)my_row * 3 + 0];
    const float py = pb[(size_t)my_row * 3 + 1];
    const float pz = pb[(size_t)my_row * 3 + 2];
    v2f a;
    a.x = (hi == 0) ? px : pz;
    a.y = (hi == 0) ? py : 1.0f;

    const v8f zero = {0.f, 0.f, 0.f, 0.f, 0.f, 0.f, 0.f, 0.f};
    v8f mind;
#pragma unroll
    for (int i = 0; i < 8; ++i) mind[i] = 3.4e38f;

    // ---- sweep all target tiles: 1 WMMA + 8 mins per 16x16 tile
#pragma unroll 4
    for (int mt = 0; mt < nt; ++mt) {
        const float* __restrict__ tp = tb + (size_t)(mt * 16 + li) * 3;
        const float tx = tp[0], ty = tp[1], tz = tp[2];
        v2f bv;
        bv.x = (hi == 0) ? (-2.0f * tx) : (-2.0f * tz);
        bv.y = (hi == 0) ? (-2.0f * ty) : (tx * tx + ty * ty + tz * tz);

        // D = A x B + 0  ->  -2 p.t + |t|^2  for a 16x16 tile
        v8f d = __builtin_amdgcn_wmma_f32_16x16x4_f32(
            /*neg_a=*/false, a, /*neg_b=*/false, bv,
            /*c_mod=*/(short)0, zero, /*reuse_a=*/false, /*reuse_b=*/false);

#pragma unroll
        for (int i = 0; i < 8; ++i) mind[i] = fminf(mind[i], d[i]);
    }

    // ---- per-row min across the 16 lanes of each half-wave
    // (xor masks 1..8 never cross the lane-16 boundary)
#pragma unroll
    for (int off = 1; off < 16; off <<= 1) {
#pragma unroll
        for (int i = 0; i < 8; ++i)
            mind[i] = fminf(mind[i], __shfl_xor(mind[i], off, 32));
    }

    // lane handles row r = (lane&7) + 8*hi ; lanes with li in 8..15 duplicate
    const int r   = (lane & 7) + 8 * hi;
    const int rn  = tile * 16 + r;
    const float qx = pb[(size_t)rn * 3 + 0];
    const float qy = pb[(size_t)rn * 3 + 1];
    const float qz = pb[(size_t)rn * 3 + 2];
    const float p2 = qx * qx + qy * qy + qz * qz;
    float d2  = mind[lane & 7] + p2;
    float dmin = sqrtf(fmaxf(d2, CHAMFER_EPS));
    float sym_c = (li < 8) ? dmin : 0.0f;    // mask duplicates

    // ---- asym (ADD) term: lanes 0-15 each handle one row of this tile
    float asym_c = 0.0f;
    if (hi == 0) {
        const float* __restrict__ tq = tb + (size_t)my_row * 3;
        const float dx = px - tq[0];
        const float dy = py - tq[1];
        const float dz = pz - tq[2];
        asym_c = sqrtf(dx * dx + dy * dy + dz * dz);
    }

    // ---- wave sum (deterministic xor tree), lane 0 writes partials
#pragma unroll
    for (int off = 1; off < 32; off <<= 1) {
        sym_c  += __shfl_xor(sym_c,  off, 32);
        asym_c += __shfl_xor(asym_c, off, 32);
    }
    if (lane == 0) {
        sym_ws [(size_t)b * nt + tile] = sym_c;
        asym_ws[(size_t)b * nt + tile] = asym_c;
    }
}

// ---------------------------------------------------------------------------
// Kernel 2: deterministic final reduction (single block, fixed tree order).
// ---------------------------------------------------------------------------
__global__ __launch_bounds__(256)
void chamfer_reduce_kernel(const float* __restrict__ sym_ws,
                           const float* __restrict__ asym_ws,
                           const float* __restrict__ sym_flag,
                           float* __restrict__ out,
                           int B, int nt, int N)
{
    __shared__ float ss[256];
    __shared__ float sa[256];
    const int t = threadIdx.x;
    float loss = 0.0f;

    for (int b = 0; b < B; ++b) {
        float s = 0.0f, a = 0.0f;
        for (int i = t; i < nt; i += 256) {
            s += sym_ws [(size_t)b * nt + i];
            a += asym_ws[(size_t)b * nt + i];
        }
        ss[t] = s; sa[t] = a;
        __syncthreads();
        for (int off = 128; off > 0; off >>= 1) {
            if (t < off) { ss[t] += ss[t + off]; sa[t] += sa[t + off]; }
            __syncthreads();
        }
        if (t == 0) {
            const float f = sym_flag[b];
            const float inv_n = 1.0f / (float)N;
            loss += f * (ss[0] * inv_n) + (1.0f - f) * (sa[0] * inv_n);
        }
        __syncthreads();   // protect ss/sa before next batch overwrites
    }
    if (t == 0) out[0] = loss / (float)B;
}

// ---------------------------------------------------------------------------
extern "C" void kernel_launch(void* const* d_in, const int* in_sizes, int n_in,
                              void* d_out, int out_size, void* d_ws, size_t ws_size,
                              hipStream_t stream)
{
    const float* pred = (const float*)d_in[0];   // [B,N,3] f32
    const float* targ = (const float*)d_in[1];   // [B,N,3] f32
    const float* flag = (const float*)d_in[2];   // [B]     f32

    const int B  = in_sizes[2];
    const int N  = in_sizes[0] / (3 * B);
    const int nt = N / 16;                       // tiles of 16 points

    float* sym_ws  = (float*)d_ws;               // [B][nt]
    float* asym_ws = sym_ws + (size_t)B * nt;    // [B][nt]

    dim3 grid(nt, B);
    chamfer_tile_kernel<<<grid, 32, 0, stream>>>(pred, targ, sym_ws, asym_ws, N, nt);
    chamfer_reduce_kernel<<<1, 256, 0, stream>>>(sym_ws, asym_ws, flag,
                                                 (float*)d_out, B, nt, N);
}